// DecoupledIntraInterMultiSequenceSelfAttention_90615220011040
// MI455X (gfx1250) — compile-verified
//
#include <hip/hip_runtime.h>
#include <hip/hip_bf16.h>
#include <math.h>

#define BB 2
#define LL 2048
#define DD 1024
#define HH 16
#define HD 64
#define SP 66   // padded LDS row stride for 64-wide tiles
#define KGEMM 1024  // K dim of every GEMM in this problem

typedef float v2f __attribute__((ext_vector_type(2)));
typedef float v8f __attribute__((ext_vector_type(8)));

static __device__ __forceinline__ v8f wmma4(v2f a, v2f b, v8f c) {
  // V_WMMA_F32_16X16X4_F32 : D = A(16x4) * B(4x16) + C(16x16)
  return __builtin_amdgcn_wmma_f32_16x16x4_f32(false, a, false, b, (short)0, c,
                                               false, false);
}

// ---------------------------------------------------------------------------
// 1) chain info: cumsum lengths, per-position cid / pos / valid
// ---------------------------------------------------------------------------
__global__ void chain_info_kernel(const int* __restrict__ lens,
                                  int* __restrict__ cid,
                                  float* __restrict__ posf,
                                  int* __restrict__ valid) {
  __shared__ int csum[LL];
  int b = blockIdx.x;
  if (threadIdx.x == 0) {
    int run = 0;
    for (int p = 0; p < LL; ++p) { run += lens[b * LL + p]; csum[p] = run; }
  }
  __syncthreads();
  int total = csum[LL - 1];
  for (int p = threadIdx.x; p < LL; p += blockDim.x) {
    int lo = 0, hi = LL;
    while (lo < hi) {                       // searchsorted(csum, p, 'right')
      int mid = (lo + hi) >> 1;
      if (csum[mid] <= p) lo = mid + 1; else hi = mid;
    }
    int c = lo;
    int cm1 = c - 1; if (cm1 < 0) cm1 = 0; if (cm1 > LL - 1) cm1 = LL - 1;
    int prev = (c > 0) ? csum[cm1] : 0;
    int vld = (p < total) ? 1 : 0;
    cid[b * LL + p] = c;
    valid[b * LL + p] = vld;
    posf[b * LL + p] = vld ? (float)(p - prev) : 0.0f;
  }
}

// ---------------------------------------------------------------------------
// 2) WMMA fp32 GEMM: C[M,N] = (A1 (+A2)) [M,1024] * W[N,1024]^T + bias[N]
//    block = 256 threads = 8 waves; wave tile 64x32 (4 M-frags x 2 N-frags);
//    waves arranged 2(M) x 4(N) -> block tile 128x128.
//    K is hardcoded (1024) so tile offsets fold into load immediates.
// ---------------------------------------------------------------------------
template <bool HAS_A2>
__global__ __launch_bounds__(256) void gemm_wmma_f32(
    const float* __restrict__ A1, const float* __restrict__ A2,
    const float* __restrict__ W, const float* __restrict__ bias,
    float* __restrict__ C, int N) {
  int lane = threadIdx.x & 31;
  int wave = threadIdx.x >> 5;
  int wm = wave >> 2;          // 0..1
  int wn = wave & 3;           // 0..3
  int tm = blockIdx.y * 128 + wm * 64;   // 4 x 16 rows
  int tn = blockIdx.x * 128 + wn * 32;   // 2 x 16 cols
  int lanelo = lane & 15;
  int khalf = (lane >> 4) << 1;   // 0 or 2
  int hi8   = (lane >> 4) << 3;   // 0 or 8

  v8f acc[4][2];
#pragma unroll
  for (int i = 0; i < 4; ++i)
#pragma unroll
    for (int j = 0; j < 2; ++j) acc[i][j] = (v8f){0,0,0,0,0,0,0,0};

  const float* pA1 = A1 + (size_t)(tm + lanelo) * KGEMM + khalf;
  const float* pA2 = HAS_A2 ? (A2 + (size_t)(tm + lanelo) * KGEMM + khalf) : nullptr;
  const float* pW  = W  + (size_t)(tn + lanelo) * KGEMM + khalf;

  for (int k0 = 0; k0 < KGEMM; k0 += 16) {
#pragma unroll
    for (int kk = 0; kk < 4; ++kk) {
      v2f a[4], b2[2];
#pragma unroll
      for (int i = 0; i < 4; ++i) {
        v2f av = *(const v2f*)(pA1 + i * 16 * KGEMM + kk * 4);
        if (HAS_A2) {
          v2f a2 = *(const v2f*)(pA2 + i * 16 * KGEMM + kk * 4);
          av.x += a2.x; av.y += a2.y;
        }
        a[i] = av;
      }
#pragma unroll
      for (int j = 0; j < 2; ++j)
        b2[j] = *(const v2f*)(pW + j * 16 * KGEMM + kk * 4);
#pragma unroll
      for (int i = 0; i < 4; ++i)
#pragma unroll
        for (int j = 0; j < 2; ++j)
          acc[i][j] = wmma4(a[i], b2[j], acc[i][j]);
    }
    pA1 += 16;
    if (HAS_A2) pA2 += 16;
    pW += 16;
  }

#pragma unroll
  for (int i = 0; i < 4; ++i)
#pragma unroll
    for (int j = 0; j < 2; ++j) {
      int col = tn + j * 16 + lanelo;
      float bv = bias[col];
#pragma unroll
      for (int v = 0; v < 8; ++v) {
        int row = tm + i * 16 + v + hi8;
        C[(size_t)row * N + col] = acc[i][j][v] + bv;
      }
    }
}

// ---------------------------------------------------------------------------
// 3) RoPE (+ q scale) applied in-place to q,k inside qkv[B,L,3,H,HD]
// ---------------------------------------------------------------------------
__global__ void rope_kernel(float* __restrict__ qkv,
                            const float* __restrict__ posf) {
  int idx = blockIdx.x * blockDim.x + threadIdx.x;   // B*L*H*32 threads
  int pair = idx & 31;
  int h = (idx >> 5) & 15;
  int l = (idx >> 9) & (LL - 1);
  int b = idx >> 20;
  float p = posf[b * LL + l];
  float invf = powf(10000.0f, -((float)(2 * pair)) / 64.0f);
  float c = cosf(p * invf);
  float s = sinf(p * invf);
  size_t qb = ((size_t)(b * LL + l) * 3 + 0) * DD + h * HD;
  size_t kb = ((size_t)(b * LL + l) * 3 + 1) * DD + h * HD;
  const float qscale = 0.125f;   // HD^-0.5
  float q1 = qkv[qb + pair], q2 = qkv[qb + pair + 32];
  qkv[qb + pair]      = qscale * (q1 * c - q2 * s);
  qkv[qb + pair + 32] = qscale * (q2 * c + q1 * s);
  float k1 = qkv[kb + pair], k2 = qkv[kb + pair + 32];
  qkv[kb + pair]      = k1 * c - k2 * s;
  qkv[kb + pair + 32] = k2 * c + k1 * s;
}

// ---------------------------------------------------------------------------
// 4) Flash attention, fp32 WMMA for QK^T and PV, online softmax in LDS.
//    grid = (B*H, L/64), block = 256 (8 waves). 64 queries x 64-key chunks.
//    wantSame=1 -> intra (same chain), 0 -> inter (different chain).
// ---------------------------------------------------------------------------
__global__ __launch_bounds__(256) void attn_kernel(
    const float* __restrict__ qkv, const int* __restrict__ cid,
    const int* __restrict__ valid, float* __restrict__ outp, int wantSame) {
  __shared__ float Ssm[64 * SP];
  __shared__ float Osm[64 * SP];
  __shared__ int qcidS[64], kcidS[64], kvalS[64];
  __shared__ float rowm[64], rowsum[64];
  __shared__ float redA[256], redB[256];

  int bh = blockIdx.x;
  int b = bh / HH, h = bh % HH;
  int q0 = blockIdx.y * 64;
  int tid = threadIdx.x;
  int lane = tid & 31;
  int wave = tid >> 5;
  int lanelo = lane & 15;
  int khalf = (lane >> 4) << 1;
  int hi8   = (lane >> 4) << 3;

  for (int i = tid; i < 64 * SP; i += 256) Osm[i] = 0.0f;
  if (tid < 64) {
    qcidS[tid] = cid[b * LL + q0 + tid];
    rowm[tid] = -INFINITY;
    rowsum[tid] = 0.0f;
  }
  __syncthreads();

  for (int kc = 0; kc < LL; kc += 64) {
    if (tid < 64) {
      kcidS[tid] = cid[b * LL + kc + tid];
      kvalS[tid] = valid[b * LL + kc + tid];
    }
    __syncthreads();

    // ---- S = Q * K^T (64x64), 16 subtiles over 8 waves ----
#pragma unroll
    for (int u = 0; u < 2; ++u) {
      int st = wave * 2 + u;
      int mt = st >> 2, nt = st & 3;
      v8f acc = (v8f){0,0,0,0,0,0,0,0};
      int qrow = q0 + mt * 16 + lanelo;
      int key  = kc + nt * 16 + lanelo;
      const float* qp = qkv + ((size_t)(b * LL + qrow) * 3 + 0) * DD + h * HD + khalf;
      const float* kp = qkv + ((size_t)(b * LL + key ) * 3 + 1) * DD + h * HD + khalf;
#pragma unroll
      for (int ks = 0; ks < 16; ++ks) {
        v2f a  = *(const v2f*)(qp + ks * 4);
        v2f bb = *(const v2f*)(kp + ks * 4);
        acc = wmma4(a, bb, acc);
      }
#pragma unroll
      for (int v = 0; v < 8; ++v) {
        int r  = mt * 16 + v + hi8;
        int cc = nt * 16 + lanelo;
        bool same = (kcidS[cc] == qcidS[r]);
        bool allowed = (kvalS[cc] != 0) && (same == (wantSame != 0));
        Ssm[r * SP + cc] = allowed ? acc[v] : -INFINITY;
      }
    }
    __syncthreads();

    // ---- online softmax update; 4 threads per row, 16 elems each ----
    {
      int row = tid >> 2, part = tid & 3;
      float* srow = &Ssm[row * SP + part * 16];
      float mx = -INFINITY;
#pragma unroll
      for (int e = 0; e < 16; ++e) mx = fmaxf(mx, srow[e]);
      redA[tid] = mx;
      __syncthreads();
      float oldm = rowm[row];
      float newmax = oldm;
#pragma unroll
      for (int qq = 0; qq < 4; ++qq) newmax = fmaxf(newmax, redA[row * 4 + qq]);
      float scale = (newmax == -INFINITY) ? 1.0f : expf(oldm - newmax);
      float psum = 0.0f;
#pragma unroll
      for (int e = 0; e < 16; ++e) {
        float sv = srow[e];
        float pv = (sv == -INFINITY) ? 0.0f : expf(sv - newmax);
        srow[e] = pv;
        psum += pv;
      }
      redB[tid] = psum;
      float* orow = &Osm[row * SP + part * 16];
#pragma unroll
      for (int e = 0; e < 16; ++e) orow[e] *= scale;
      __syncthreads();
      if (part == 0) {
        rowsum[row] = rowsum[row] * scale +
                      redB[row * 4] + redB[row * 4 + 1] +
                      redB[row * 4 + 2] + redB[row * 4 + 3];
        rowm[row] = newmax;
      }
      __syncthreads();
    }

    // ---- O += P * V  (64x64 += 64x64 * 64x64) ----
#pragma unroll
    for (int u = 0; u < 2; ++u) {
      int st = wave * 2 + u;
      int mt = st >> 2, nt = st & 3;
      v8f acc;
#pragma unroll
      for (int v = 0; v < 8; ++v)
        acc[v] = Osm[(mt * 16 + v + hi8) * SP + nt * 16 + lanelo];
      const float* vbase = qkv + ((size_t)(b * LL + kc) * 3 + 2) * DD +
                           h * HD + (nt * 16 + lanelo);
#pragma unroll
      for (int ks = 0; ks < 16; ++ks) {
        int kd = ks * 4 + khalf;
        v2f a = *(const v2f*)(&Ssm[(mt * 16 + lanelo) * SP + kd]);
        v2f bb;
        bb.x = vbase[(size_t)kd * (3 * DD)];
        bb.y = vbase[(size_t)(kd + 1) * (3 * DD)];
        acc = wmma4(a, bb, acc);
      }
#pragma unroll
      for (int v = 0; v < 8; ++v)
        Osm[(mt * 16 + v + hi8) * SP + nt * 16 + lanelo] = acc[v];
    }
    __syncthreads();
  }

  // ---- normalize (sum==0 -> 0, the NaN->0 path) and store ----
  for (int i = tid; i < 64 * 64; i += 256) {
    int r = i >> 6, dcol = i & 63;
    float sm = rowsum[r];
    float val = (sm > 0.0f) ? Osm[r * SP + dcol] / sm : 0.0f;
    outp[(size_t)(b * LL + q0 + r) * DD + h * HD + dcol] = val;
  }
}

// ---------------------------------------------------------------------------
// launch
// ---------------------------------------------------------------------------
extern "C" void kernel_launch(void* const* d_in, const int* in_sizes, int n_in,
                              void* d_out, int out_size, void* d_ws, size_t ws_size,
                              hipStream_t stream) {
  const float* x    = (const float*)d_in[0];
  const int*   lens = (const int*)  d_in[1];
  const float* Wi   = (const float*)d_in[2];
  const float* bi   = (const float*)d_in[3];
  const float* Wt   = (const float*)d_in[4];
  const float* bt   = (const float*)d_in[5];
  const float* Wo   = (const float*)d_in[6];
  const float* bo   = (const float*)d_in[7];
  float* out = (float*)d_out;

  char* ws = (char*)d_ws;
  int*   cid   = (int*)  (ws + 0);
  float* posf  = (float*)(ws + 32768);
  int*   vldb  = (int*)  (ws + 65536);
  float* qkv   = (float*)(ws + 131072);                       // 4096*3072 f32
  float* o_in  = (float*)(ws + 131072 + (size_t)50331648);    // 4096*1024 f32
  float* o_ex  = (float*)(ws + 131072 + (size_t)50331648 + 16777216);

  const int M = BB * LL;          // 4096
  const int NQKV = 3 * DD;        // 3072

  chain_info_kernel<<<BB, 256, 0, stream>>>(lens, cid, posf, vldb);

  dim3 gQKV(NQKV / 128, M / 128);   // (24, 32)
  dim3 gOUT(DD / 128, M / 128);     // (8, 32)
  dim3 gATT(BB * HH, LL / 64);      // (32, 32)
  int ropeBlocks = (BB * LL * HH * 32) / 256;

  // intra branch
  gemm_wmma_f32<false><<<gQKV, 256, 0, stream>>>(x, nullptr, Wi, bi, qkv, NQKV);
  rope_kernel<<<ropeBlocks, 256, 0, stream>>>(qkv, posf);
  attn_kernel<<<gATT, 256, 0, stream>>>(qkv, cid, vldb, o_in, 1);

  // inter branch (reuse qkv buffer)
  gemm_wmma_f32<false><<<gQKV, 256, 0, stream>>>(x, nullptr, Wt, bt, qkv, NQKV);
  rope_kernel<<<ropeBlocks, 256, 0, stream>>>(qkv, posf);
  attn_kernel<<<gATT, 256, 0, stream>>>(qkv, cid, vldb, o_ex, 0);

  // final projection with fused sum of the two branch outputs
  gemm_wmma_f32<true><<<gOUT, 256, 0, stream>>>(o_in, o_ex, Wo, bo, out, DD);
}